// BlackBoxV3_14877766713680
// MI455X (gfx1250) — compile-verified
//
#include <hip/hip_runtime.h>
#include <math.h>

#define D        128
#define BATCH    4
#define SEQ      2048
#define VOCAB_SZ 32000
#define N_INNER  4
#define NTOK     (BATCH * SEQ)
#define PITCH    132    // LDS row pitch in floats (bank-conflict padding, 8B aligned)
#define CPITCH   260

typedef float v2f __attribute__((ext_vector_type(2)));
typedef float v8f __attribute__((ext_vector_type(8)));

__device__ __forceinline__ float gelu_exact(float x) {
    return 0.5f * x * (1.0f + erff(x * 0.70710678118654752f));
}
__device__ __forceinline__ float sigmoid_f(float x) {
    return 1.0f / (1.0f + expf(-x));
}

// ---------------------------------------------------------------------------
// Kernel 1: gather embedding + precompute mod_all = tok @ mod_w.T and
// tokg_all = tok @ gate_w[:, D:].T for every token (fully parallel).
// ---------------------------------------------------------------------------
__global__ void __launch_bounds__(D)
tok_precompute(const int* __restrict__ ids,
               const float* __restrict__ embed_w,
               const float* __restrict__ gate_w,
               const float* __restrict__ mod_w,
               float* __restrict__ mod_all,
               float* __restrict__ tokg_all)
{
    __shared__ float tok[D];
    const int t  = blockIdx.x;      // flat token index = b*SEQ + n
    const int j  = threadIdx.x;     // output feature
    const int id = ids[t];
    tok[j] = embed_w[id * D + j];
    __syncthreads();

    const float* mw = mod_w  + j * D;
    const float* gw = gate_w + j * (2 * D) + D;   // token half of gate weight
    float m = 0.0f, g = 0.0f;
#pragma unroll 8
    for (int d = 0; d < D; ++d) {
        const float td = tok[d];
        m = fmaf(td, mw[d], m);
        g = fmaf(td, gw[d], g);
    }
    mod_all [t * D + j] = m;
    tokg_all[t * D + j] = g;
}

// ---------------------------------------------------------------------------
// Kernel 2: the sequential recurrence. One workgroup, 8 wave32s.
// Weights (W and gate_w state-half) + state live entirely in LDS (~143 KB of
// the 320 KB WGP pool). Wave w owns output columns [16w, 16w+16). Batch rows
// 0..3 ride in the M dimension of 16x16x4 f32 WMMA tiles; rows 4..15 stay 0.
// Each matmul uses two independent accumulation chains for WMMA pipelining;
// per-step mod/tok_gate loads are software-pipelined one step ahead.
// ---------------------------------------------------------------------------
__global__ void __launch_bounds__(256, 1)
recurrence(const float* __restrict__ W,
           const float* __restrict__ gate_w,
           const float* __restrict__ gate_b,
           const float* __restrict__ mod_all,
           const float* __restrict__ tokg_all,
           float* __restrict__ states)
{
    __shared__ float sW[D * PITCH];    // sW[j][k] = W[j][k]
    __shared__ float sG[D * PITCH];    // sG[j][k] = gate_w[j][k]   (state half)
    __shared__ float sS[16 * PITCH];   // state, rows 4..15 == 0

    const int tid  = threadIdx.x;
    const int wave = tid >> 5;
    const int lane = tid & 31;
    const int half = lane >> 4;        // K-pair selector for A/B fragments
    const int l15  = lane & 15;
    const int jcol = wave * 16 + l15;  // this lane's output column (N index)

    for (int i = tid; i < D * D; i += 256) {
        const int j = i >> 7, k = i & (D - 1);
        sW[j * PITCH + k] = W[i];
        sG[j * PITCH + k] = gate_w[j * (2 * D) + k];
    }
    for (int i = tid; i < 16 * PITCH; i += 256) sS[i] = 0.0f;
    __syncthreads();

    const float gb = gate_b[jcol];

    v8f st;                            // state tile in C layout (persistent)
#pragma unroll
    for (int r = 0; r < 8; ++r) st[r] = 0.0f;

    // prime the mod/tok_gate pipeline with step 0
    float cmod[4], ctg[4];
#pragma unroll
    for (int b = 0; b < 4; ++b) {
        const int off = (b * SEQ) * D + jcol;
        cmod[b] = mod_all[off];
        ctg[b]  = tokg_all[off] + gb;
    }

    for (int step = 0; step < SEQ; ++step) {
        // issue next step's loads now; latency hides under the inner loop
        float nmod[4], ntg[4];
        const int s1 = (step + 1 < SEQ) ? (step + 1) : step;
#pragma unroll
        for (int b = 0; b < 4; ++b) {
            const int off = (b * SEQ + s1) * D + jcol;
            nmod[b] = mod_all[off];
            ntg[b]  = tokg_all[off] + gb;
        }

        for (int it = 0; it < N_INNER; ++it) {
            v8f cw0, cw1, cg0, cg1;    // two chains per matmul
#pragma unroll
            for (int r = 0; r < 8; ++r) {
                cw0[r] = 0.0f; cw1[r] = 0.0f; cg0[r] = 0.0f; cg1[r] = 0.0f;
            }
            if (half == 0) {           // rows M=0..3 get mod / tok_gate init
#pragma unroll
                for (int r = 0; r < 4; ++r) { cw0[r] = cmod[r]; cg0[r] = ctg[r]; }
            }

#pragma unroll 4
            for (int k0 = 0; k0 < D; k0 += 8) {
                const int kk = k0 + half * 2;
                const v2f a0  = *(const v2f*)&sS[l15  * PITCH + kk];
                const v2f bw0 = *(const v2f*)&sW[jcol * PITCH + kk];
                const v2f bg0 = *(const v2f*)&sG[jcol * PITCH + kk];
                const v2f a1  = *(const v2f*)&sS[l15  * PITCH + kk + 4];
                const v2f bw1 = *(const v2f*)&sW[jcol * PITCH + kk + 4];
                const v2f bg1 = *(const v2f*)&sG[jcol * PITCH + kk + 4];
                cw0 = __builtin_amdgcn_wmma_f32_16x16x4_f32(
                        false, a0, false, bw0, (short)0, cw0, false, false);
                cg0 = __builtin_amdgcn_wmma_f32_16x16x4_f32(
                        false, a0, false, bg0, (short)0, cg0, false, false);
                cw1 = __builtin_amdgcn_wmma_f32_16x16x4_f32(
                        false, a1, false, bw1, (short)0, cw1, false, false);
                cg1 = __builtin_amdgcn_wmma_f32_16x16x4_f32(
                        false, a1, false, bg1, (short)0, cg1, false, false);
            }

            const v8f cw = cw0 + cw1;
            const v8f cg = cg0 + cg1;
#pragma unroll
            for (int r = 0; r < 8; ++r) {
                const float ns = gelu_exact(cw[r]);
                const float g  = sigmoid_f(cg[r]);
                st[r] = fmaf(g, ns - st[r], st[r]);   // g*ns + (1-g)*st
            }

            __syncthreads();           // everyone done reading old state
            if (half == 0) {
#pragma unroll
                for (int r = 0; r < 4; ++r) sS[r * PITCH + jcol] = st[r];
            }
            __syncthreads();           // new state visible to all waves
        }

        if (half == 0) {
#pragma unroll
            for (int r = 0; r < 4; ++r)
                states[(size_t)(r * SEQ + step) * D + jcol] = st[r];
        }

#pragma unroll
        for (int b = 0; b < 4; ++b) { cmod[b] = nmod[b]; ctg[b] = ntg[b]; }
    }
}

// ---------------------------------------------------------------------------
// Kernel 3: logits = states @ out_w.T + out_b, f32 WMMA.
// Block: 32 tokens x 256 vocab. 8 waves, each owning two 16-col vocab tiles
// across both 16-token M-tiles (4 accumulator chains). Each out_w B-fragment
// feeds two WMMAs, halving L2 traffic vs a 16-token tile. C is transposed
// through LDS so the 1.05 GB output is written fully coalesced (1 KB rows).
// ---------------------------------------------------------------------------
__global__ void __launch_bounds__(256)
logits_kernel(const float* __restrict__ states,
              const float* __restrict__ out_w,
              const float* __restrict__ out_b,
              float* __restrict__ logits)
{
    __shared__ float sA[32 * PITCH];     // 32 tokens x 128 K
    __shared__ float sC[32 * CPITCH];    // 32 tokens x 256 vocab staging

    const int tid  = threadIdx.x;
    const int wave = tid >> 5;
    const int lane = tid & 31;
    const int half = lane >> 4;
    const int l15  = lane & 15;

    const int tok0 = blockIdx.y * 32;
    const int vblk = blockIdx.x * 256;
    const int v0   = vblk + wave * 32;

    for (int i = tid; i < 32 * D; i += 256) {
        const int m = i >> 7, k = i & (D - 1);
        sA[m * PITCH + k] = states[(size_t)(tok0 + m) * D + k];
    }
    __syncthreads();

    const int va = v0 + l15;
    const int vb = v0 + 16 + l15;
    const float biasa = out_b[va];
    const float biasb = out_b[vb];

    v8f c00, c01, c10, c11;              // [Mtile][Ntile]
#pragma unroll
    for (int r = 0; r < 8; ++r) {
        c00[r] = biasa; c01[r] = biasb; c10[r] = biasa; c11[r] = biasb;
    }

    const float* wa = out_w + (size_t)va * D;
    const float* wb = out_w + (size_t)vb * D;
#pragma unroll 8
    for (int k0 = 0; k0 < D; k0 += 4) {
        const int kk = k0 + half * 2;
        const v2f aL = *(const v2f*)&sA[l15 * PITCH + kk];           // tokens 0..15
        const v2f aH = *(const v2f*)&sA[(16 + l15) * PITCH + kk];    // tokens 16..31
        const v2f b0 = *(const v2f*)&wa[kk];
        const v2f b1 = *(const v2f*)&wb[kk];
        c00 = __builtin_amdgcn_wmma_f32_16x16x4_f32(
                false, aL, false, b0, (short)0, c00, false, false);
        c01 = __builtin_amdgcn_wmma_f32_16x16x4_f32(
                false, aL, false, b1, (short)0, c01, false, false);
        c10 = __builtin_amdgcn_wmma_f32_16x16x4_f32(
                false, aH, false, b0, (short)0, c10, false, false);
        c11 = __builtin_amdgcn_wmma_f32_16x16x4_f32(
                false, aH, false, b1, (short)0, c11, false, false);
    }

#pragma unroll
    for (int r = 0; r < 8; ++r) {
        const int mL = r + half * 8;
        const int mH = 16 + mL;
        sC[mL * CPITCH + wave * 32 + l15]      = c00[r];
        sC[mL * CPITCH + wave * 32 + 16 + l15] = c01[r];
        sC[mH * CPITCH + wave * 32 + l15]      = c10[r];
        sC[mH * CPITCH + wave * 32 + 16 + l15] = c11[r];
    }
    __syncthreads();

#pragma unroll 4
    for (int m = 0; m < 32; ++m) {
        logits[(size_t)(tok0 + m) * VOCAB_SZ + vblk + tid] = sC[m * CPITCH + tid];
    }
}

// ---------------------------------------------------------------------------
extern "C" void kernel_launch(void* const* d_in, const int* in_sizes, int n_in,
                              void* d_out, int out_size, void* d_ws, size_t ws_size,
                              hipStream_t stream)
{
    (void)in_sizes; (void)n_in; (void)out_size; (void)ws_size;

    const int*   ids     = (const int*)  d_in[0];
    const float* embed_w = (const float*)d_in[1];
    const float* W       = (const float*)d_in[2];
    const float* gate_w  = (const float*)d_in[3];
    const float* gate_b  = (const float*)d_in[4];
    const float* mod_w   = (const float*)d_in[5];
    const float* out_w   = (const float*)d_in[6];
    const float* out_b   = (const float*)d_in[7];
    float*       logits  = (float*)d_out;

    char*  ws       = (char*)d_ws;
    float* mod_all  = (float*)(ws);
    float* tokg_all = (float*)(ws + 1 * (size_t)NTOK * D * sizeof(float));
    float* states   = (float*)(ws + 2 * (size_t)NTOK * D * sizeof(float));

    tok_precompute<<<NTOK, D, 0, stream>>>(ids, embed_w, gate_w, mod_w,
                                           mod_all, tokg_all);

    recurrence<<<1, 256, 0, stream>>>(W, gate_w, gate_b,
                                      mod_all, tokg_all, states);

    dim3 grid(VOCAB_SZ / 256, NTOK / 32);
    logits_kernel<<<grid, 256, 0, stream>>>(states, out_w, out_b, logits);
}